// QKRotationWrapper_46926812676137
// MI455X (gfx1250) — compile-verified
//
#include <hip/hip_runtime.h>
#include <hip/hip_bf16.h>

// FWHT(128) as H128 * X^T GEMM on the WMMA pipe (fp32, exact vs reference),
// + per-row mixed 4/8-bit fake-quant epilogue for K.
//
// H128[p][q] = (-1)^popcount(p&q)  (Sylvester ordering, matches reference fwht)

typedef float v2f __attribute__((ext_vector_type(2)));
typedef float v4f __attribute__((ext_vector_type(4)));
typedef float v8f __attribute__((ext_vector_type(8)));

#define ROW_DWORDS 128
#define LDS_STRIDE 132          // 128 + 4 pad dwords: conflict-free strided B gather
#define WAVES_PER_BLOCK 4

template <bool QUANT>
__global__ __launch_bounds__(WAVES_PER_BLOCK * 32)
void fwht_hadamard_wmma(const float* __restrict__ in, float* __restrict__ out,
                        int num_tiles /* tiles of 16 rows */) {
    __shared__ float lds[WAVES_PER_BLOCK * 16 * LDS_STRIDE];
    const int lane = threadIdx.x & 31;
    const int wave = threadIdx.x >> 5;
    float* wlds = &lds[wave * 16 * LDS_STRIDE];

    const int m   = lane & 15;   // A-row M / B-col N / D-col N for this lane
    const int hlf = lane >> 4;   // lane half selects K (A/B) and M-half (D)
    const float inv_sqrt_d = 0.08838834764831845f;  // 1/sqrt(128)

    for (int tile = blockIdx.x * WAVES_PER_BLOCK + wave; tile < num_tiles;
         tile += gridDim.x * WAVES_PER_BLOCK) {
        const float* src = in  + (size_t)tile * (16 * ROW_DWORDS);
        float*       dst = out + (size_t)tile * (16 * ROW_DWORDS);

        // ---- stage 16 rows into this wave's private LDS slab (coalesced b128) ----
        #pragma unroll
        for (int i = 0; i < 16; ++i) {
            v4f v = __builtin_nontemporal_load(((const v4f*)(src + i * ROW_DWORDS)) + lane);
            *(v4f*)(wlds + i * LDS_STRIDE + lane * 4) = v;
        }

        // ---- Y^T = H128 * X^T via v_wmma_f32_16x16x4_f32 ----
        // 8 output row-blocks (pb) x 32 K-chunks (kc) of 4.
        v8f c[8];
        #pragma unroll
        for (int pb = 0; pb < 8; ++pb) { v8f z = {0,0,0,0,0,0,0,0}; c[pb] = z; }

        #pragma unroll
        for (int kc = 0; kc < 32; ++kc) {
            // B[k][n] = x_n[4*kc + k]; this lane supplies n=m, k = 2*hlf + {0,1}
            v2f b = *(const v2f*)(wlds + m * LDS_STRIDE + kc * 4 + hlf * 2);
            v2f bn = -b;

            // A[mrow][k] = H128[16*pb + m][4*kc + k] = (+/-1)
            // sign = parity(m & (q&15)) XOR parity(pb & (q>>4)); q>>4 == kc>>2 here.
            const int qlow = ((kc & 3) << 2) + (hlf << 1);
            unsigned p0 = (unsigned)__popc(m & qlow)       & 1u;
            unsigned p1 = (unsigned)__popc(m & (qlow + 1)) & 1u;
            v2f a;
            a.x = __int_as_float((int)(0x3F800000u | (p0 << 31)));
            a.y = __int_as_float((int)(0x3F800000u | (p1 << 31)));

            const int khi = kc >> 2;
            #pragma unroll
            for (int pb = 0; pb < 8; ++pb) {
                // scalar part of the Hadamard sign: fold as +/-B (compile-time select)
                const bool flip = (__popc(pb & khi) & 1) != 0;
                c[pb] = __builtin_amdgcn_wmma_f32_16x16x4_f32(
                    false, a, false, (flip ? bn : b), (short)0, c[pb], false, false);
            }
        }

        // ---- D -> LDS: lane's column N = data row m; VGPR v = output col ----
        // D layout: lanes 0-15 hold M=v, lanes 16-31 hold M=v+8  => col p = 16*pb + 8*hlf + v
        #pragma unroll
        for (int pb = 0; pb < 8; ++pb) {
            float* wp = wlds + m * LDS_STRIDE + pb * 16 + hlf * 8;
            #pragma unroll
            for (int v = 0; v < 8; ++v) wp[v] = c[pb][v];
        }

        // ---- epilogue: scale (+ K fake-quant), coalesced b128 store ----
        #pragma unroll
        for (int row = 0; row < 16; ++row) {
            v4f y = *(const v4f*)(wlds + row * LDS_STRIDE + lane * 4);
            y.x *= inv_sqrt_d; y.y *= inv_sqrt_d; y.z *= inv_sqrt_d; y.w *= inv_sqrt_d;
            if (QUANT) {
                // lane holds cols [4*lane, 4*lane+3]: lanes 0-23 -> cols 0..95 (4-bit),
                // lanes 24-31 -> cols 96..127 (8-bit)
                float m4 = fmaxf(fmaxf(fabsf(y.x), fabsf(y.y)),
                                 fmaxf(fabsf(y.z), fabsf(y.w)));
                float mlow  = (lane < 24) ? m4 : 0.0f;
                float mhigh = (lane < 24) ? 0.0f : m4;
                #pragma unroll
                for (int s = 16; s >= 1; s >>= 1) {
                    mlow  = fmaxf(mlow,  __shfl_xor(mlow,  s, 32));
                    mhigh = fmaxf(mhigh, __shfl_xor(mhigh, s, 32));
                }
                const bool  low   = (lane < 24);
                const float maxq  = low ? 7.0f   : 127.0f;
                const float qmin  = low ? -8.0f  : -128.0f;
                const float xmax  = (low ? mlow : mhigh) * 0.95f;
                const float scale = (xmax > 0.0f) ? (xmax / maxq) : 1.0f;
                y.x = fminf(fmaxf(rintf(y.x / scale), qmin), maxq) * scale;
                y.y = fminf(fmaxf(rintf(y.y / scale), qmin), maxq) * scale;
                y.z = fminf(fmaxf(rintf(y.z / scale), qmin), maxq) * scale;
                y.w = fminf(fmaxf(rintf(y.w / scale), qmin), maxq) * scale;
            }
            __builtin_nontemporal_store(y, ((v4f*)(dst + row * ROW_DWORDS)) + lane);
        }
    }
}

extern "C" void kernel_launch(void* const* d_in, const int* in_sizes, int n_in,
                              void* d_out, int out_size, void* d_ws, size_t ws_size,
                              hipStream_t stream) {
    (void)n_in; (void)out_size; (void)d_ws; (void)ws_size;
    const float* q = (const float*)d_in[0];   // (2,32,4096,128) fp32
    const float* k = (const float*)d_in[1];   // (2,8,4096,128)  fp32
    float* q_out = (float*)d_out;
    float* k_out = q_out + (size_t)in_sizes[0];

    const int q_tiles = in_sizes[0] / (16 * ROW_DWORDS);  // 16384
    const int k_tiles = in_sizes[1] / (16 * ROW_DWORDS);  // 4096

    const int q_blocks = (q_tiles + WAVES_PER_BLOCK - 1) / WAVES_PER_BLOCK;
    const int k_blocks = (k_tiles + WAVES_PER_BLOCK - 1) / WAVES_PER_BLOCK;

    fwht_hadamard_wmma<false><<<q_blocks, WAVES_PER_BLOCK * 32, 0, stream>>>(q, q_out, q_tiles);
    fwht_hadamard_wmma<true ><<<k_blocks, WAVES_PER_BLOCK * 32, 0, stream>>>(k, k_out, k_tiles);
}